// StableZeroDiv_16561393894029
// MI455X (gfx1250) — compile-verified
//
#include <hip/hip_runtime.h>

// out = x * (1/y where y != 0 else 0), fp32 elementwise over 64*1048576 elems.
// HBM-bound streaming kernel: b128 non-temporal loads/stores, 4x float4 per
// thread for memory-level parallelism, wave32-native 256-thread blocks.

typedef float v4f __attribute__((ext_vector_type(4)));

__device__ __forceinline__ v4f safe_div4(v4f xv, v4f yv) {
    v4f r;
#pragma unroll
    for (int k = 0; k < 4; ++k) {
        float yk  = yv[k];
        float inv = 1.0f / yk;                    // IEEE fp32 divide
        r[k] = (yk != 0.0f) ? xv[k] * inv : 0.0f; // branchless select
    }
    return r;
}

__global__ __launch_bounds__(256) void StableZeroDiv_16561393894029_kernel(
    const float* __restrict__ x, const float* __restrict__ y,
    float* __restrict__ out, long long n) {
    const long long n4     = n >> 2;                                  // float4 count
    const long long tid    = (long long)blockIdx.x * blockDim.x + threadIdx.x;
    const long long stride = (long long)gridDim.x * blockDim.x;

    const v4f* __restrict__ x4 = (const v4f*)x;
    const v4f* __restrict__ y4 = (const v4f*)y;
    v4f* __restrict__ o4       = (v4f*)out;

    // 4 independent float4 elements per thread, strided so each batch of
    // loads stays fully coalesced across the wave. Issue all loads first
    // (8 x global_load_b128 NT in flight), then compute + NT store.
    const long long i0 = tid;
    const long long i1 = tid + stride;
    const long long i2 = tid + 2 * stride;
    const long long i3 = tid + 3 * stride;

    v4f a0, b0, a1, b1, a2, b2, a3, b3;
    const bool p0 = i0 < n4, p1 = i1 < n4, p2 = i2 < n4, p3 = i3 < n4;

    if (p0) { a0 = __builtin_nontemporal_load(&x4[i0]); b0 = __builtin_nontemporal_load(&y4[i0]); }
    if (p1) { a1 = __builtin_nontemporal_load(&x4[i1]); b1 = __builtin_nontemporal_load(&y4[i1]); }
    if (p2) { a2 = __builtin_nontemporal_load(&x4[i2]); b2 = __builtin_nontemporal_load(&y4[i2]); }
    if (p3) { a3 = __builtin_nontemporal_load(&x4[i3]); b3 = __builtin_nontemporal_load(&y4[i3]); }

    if (p0) __builtin_nontemporal_store(safe_div4(a0, b0), &o4[i0]);
    if (p1) __builtin_nontemporal_store(safe_div4(a1, b1), &o4[i1]);
    if (p2) __builtin_nontemporal_store(safe_div4(a2, b2), &o4[i2]);
    if (p3) __builtin_nontemporal_store(safe_div4(a3, b3), &o4[i3]);

    // Scalar tail for n not divisible by 4 (not hit for B*N = 2^26, but safe).
    const long long tail = n4 << 2;
    if (blockIdx.x == 0) {
        for (long long j = tail + threadIdx.x; j < n; j += blockDim.x) {
            float yk  = y[j];
            float inv = 1.0f / yk;
            out[j] = (yk != 0.0f) ? x[j] * inv : 0.0f;
        }
    }
}

extern "C" void kernel_launch(void* const* d_in, const int* in_sizes, int n_in,
                              void* d_out, int out_size, void* d_ws, size_t ws_size,
                              hipStream_t stream) {
    (void)n_in; (void)out_size; (void)d_ws; (void)ws_size;
    const float* x = (const float*)d_in[0];
    const float* y = (const float*)d_in[1];
    float* out     = (float*)d_out;

    const long long n  = (long long)in_sizes[0];   // 64 * 1048576 = 67,108,864
    const long long n4 = n >> 2;                   // float4 elements
    // 4 float4 (16 floats) per thread
    const long long threads_needed = (n4 + 3) / 4;
    int blocks = (int)((threads_needed + 255) / 256);
    if (blocks < 1) blocks = 1;

    StableZeroDiv_16561393894029_kernel<<<blocks, 256, 0, stream>>>(x, y, out, n);
}